// CRFRNN_43705587204158
// MI455X (gfx1250) — compile-verified
//
#include <hip/hip_runtime.h>

typedef float v2f __attribute__((ext_vector_type(2)));
typedef float v4f __attribute__((ext_vector_type(4)));
typedef float v8f __attribute__((ext_vector_type(8)));

#define CN    16
#define HH    64
#define WW    1024
#define HWN   65536          // H*W
#define BN    4
#define NTAP  14
#define TH    4              // tile rows per workgroup
#define TWID  64             // tile cols per workgroup
#define HTH   (TH + 2)       // halo rows  (KH=3 -> +-1)
#define HTW   (TWID + 4)     // halo cols  (KW=5 -> +-2)
#define PLANE (HTH * HTW)    // 408 halo pixels
#define STR   20             // per-pixel LDS stride (16 ch + 4 pad) -> 64-bank tiling

#define TAP_TABLES                                                        \
  const int DHt[NTAP] = {-1,-1,-1,-1,-1, 0, 0, 0, 0, 1, 1, 1, 1, 1};      \
  const int DWt[NTAP] = {-2,-1, 0, 1, 2,-2,-1, 1, 2,-2,-1, 0, 1, 2};

// ---------------------------------------------------------------------------
// Kernel 1: precompute bilateral kernel_beta[b][tap][h][w] (loop-invariant
// across the 3 CRF iterations).  1/(2*0.015^2) = 2222.2222
// ---------------------------------------------------------------------------
__global__ __launch_bounds__(256) void crf_beta_kernel(
    const float* __restrict__ xyz, float* __restrict__ kbeta) {
  const int g  = blockIdx.x * 256 + threadIdx.x;   // 0 .. B*H*W-1
  const int b  = g >> 16;
  const int hw = g & (HWN - 1);
  const int y  = hw >> 10;
  const int x  = hw & (WW - 1);
  const float* xb = xyz + (size_t)b * 3 * HWN;
  const float cx = xb[hw], cy = xb[HWN + hw], cz = xb[2 * HWN + hw];
  TAP_TABLES
#pragma unroll
  for (int t = 0; t < NTAP; ++t) {
    const int yy = y + DHt[t];
    const int xx = x + DWt[t];
    float nx = 0.f, ny = 0.f, nz = 0.f;        // zero-padded tensor (ref semantics)
    if (yy >= 0 && yy < HH && xx >= 0 && xx < WW) {
      const int np = yy * WW + xx;
      nx = xb[np]; ny = xb[HWN + np]; nz = xb[2 * HWN + np];
    }
    const float dx = nx - cx, dy = ny - cy, dz = nz - cz;
    const float d2 = dx * dx + dy * dy + dz * dz;
    kbeta[((size_t)b * NTAP + t) * HWN + hw] = __expf(-d2 * 2222.2222222f);
  }
}

// ---------------------------------------------------------------------------
// Kernel 2: one fused CRF mean-field iteration.
//   softmax -> (shared) 14-tap Gaussian conv s[] -> bilateral term e[] ->
//   weighted = s*(w_smooth + w_app*mask*e) -> pairwise = W_compat @ weighted
//   (V_WMMA_F32_16X16X4_F32, 4 chained K-chunks) -> Q = unary - pairwise
// LDS is pixel-major with stride-20 padding: per-lane channel pairs are one
// conflict-free ds_load_b64, softmax writes are 4x ds_store_b128 per pixel.
// ---------------------------------------------------------------------------
__global__ __launch_bounds__(256) void crf_iter_kernel(
    const float* __restrict__ Qin,   const float* __restrict__ unary,
    const float* __restrict__ mask,  const float* __restrict__ kbeta,
    const float* __restrict__ w_app, const float* __restrict__ w_smooth,
    const float* __restrict__ Wc,    float* __restrict__ Qout) {
  __shared__ float Pl[PLANE * STR];  // softmax(Q): [halo pixel][channel(+pad)]
  __shared__ float Ml[PLANE];        // mask tile (halo incl.)

  const int tid  = threadIdx.x;
  const int lane = tid & 31;
  const int wave = tid >> 5;         // 0..7
  const int half = (lane >> 4) & 1;  // WMMA half-wave
  const int idx  = lane & 15;        // WMMA row/col index

  const int b  = blockIdx.z;
  const int y0 = blockIdx.y * TH;
  const int x0 = blockIdx.x * TWID;

  TAP_TABLES

  // ---- stage 1: channel softmax into LDS over the haloed tile ----
  const float* qb = Qin + (size_t)b * CN * HWN;
  for (int p = tid; p < PLANE; p += 256) {
    const int rr = p / HTW;
    const int xx = p - rr * HTW;
    const int y  = y0 + rr - 1;
    const int x  = x0 + xx - 2;
    float m = 0.0f;
    float e[CN];
    if (y >= 0 && y < HH && x >= 0 && x < WW) {
      const int pix = y * WW + x;
      m = mask[(size_t)b * HWN + pix];
      float v[CN];
      float mx = -3.402823466e38f;
#pragma unroll
      for (int c = 0; c < CN; ++c) { v[c] = qb[(size_t)c * HWN + pix]; mx = fmaxf(mx, v[c]); }
      float s = 0.0f;
#pragma unroll
      for (int c = 0; c < CN; ++c) { e[c] = __expf(v[c] - mx); s += e[c]; }
      const float inv = 1.0f / s;
#pragma unroll
      for (int c = 0; c < CN; ++c) e[c] *= inv;
    } else {
#pragma unroll
      for (int c = 0; c < CN; ++c) e[c] = 0.0f;   // zero padding outside image
    }
    Ml[p] = m;
    v4f* dst = (v4f*)&Pl[p * STR];               // 80B/pixel -> 16B aligned
#pragma unroll
    for (int j = 0; j < 4; ++j) {
      v4f q = {e[4 * j], e[4 * j + 1], e[4 * j + 2], e[4 * j + 3]};
      dst[j] = q;                                // ds_store_b128
    }
  }

  // ---- per-lane constants (WMMA operand layout) ----
  // B-operand channel pair for chunk k: c = 4k + 2*half + {0,1}
  v2f A[4];                    // W_compat, A-matrix 16x4 chunks (M=idx, K=chunk)
  v2f walv[4], wslv[4];
#pragma unroll
  for (int k = 0; k < 4; ++k) {
    const int c0 = 4 * k + 2 * half;
    A[k].x   = Wc[idx * 16 + c0];
    A[k].y   = Wc[idx * 16 + c0 + 1];
    walv[k]  = *(const v2f*)&w_app[c0];
    wslv[k]  = *(const v2f*)&w_smooth[c0];
  }

  // spatial Gaussian weights (theta_gamma == theta_alpha == 0.9 -> one conv)
  float wsp[NTAP];
#pragma unroll
  for (int t = 0; t < NTAP; ++t)
    wsp[t] = __expf(-(float)(DHt[t] * DHt[t] + DWt[t] * DWt[t]) * 0.6172839506f);

  __syncthreads();

  // ---- stage 2: message passing + WMMA channel mix, 16 pixels per group ----
  for (int G = wave; G < 16; G += 8) {          // uniform per wave: EXEC all-1s
    const int tr = G >> 2;
    const int gc = G & 3;
    const int y  = y0 + tr;
    const int x  = x0 + gc * 16 + idx;
    const int hbase = (tr + 1) * HTW + (gc * 16 + idx + 2);   // halo pixel index

    const float mc = Ml[hbase];                 // center mask
    const float* kbp = kbeta + (size_t)b * NTAP * HWN + y * WW + x;
    float kbm[NTAP];                            // kernel_beta * neighbor mask
#pragma unroll
    for (int t = 0; t < NTAP; ++t)
      kbm[t] = kbp[(size_t)t * HWN] * Ml[hbase + DHt[t] * HTW + DWt[t]];

    v2f sv[4], ev[4];
#pragma unroll
    for (int k = 0; k < 4; ++k) { sv[k] = (v2f){0.f, 0.f}; ev[k] = (v2f){0.f, 0.f}; }
#pragma unroll
    for (int t = 0; t < NTAP; ++t) {
      const int nbo = (hbase + DHt[t] * HTW + DWt[t]) * STR + 2 * half;
      const float wt = wsp[t];
      const float kt = kbm[t];
#pragma unroll
      for (int k = 0; k < 4; ++k) {
        const v2f pv = *(const v2f*)&Pl[nbo + 4 * k];   // ds_load_b64, bank-tiled
        sv[k] = pv * wt + sv[k];                 // k_smooth == exp_smooth
        ev[k] = pv * kt + ev[k];                 // bilateral gather
      }
    }

    // pairwise = W_compat(16x16) @ weighted(16 x 16 pixels) via 4 chained
    // V_WMMA_F32_16X16X4_F32 (full f32 precision, matches reference)
    v8f acc = {0.f, 0.f, 0.f, 0.f, 0.f, 0.f, 0.f, 0.f};
#pragma unroll
    for (int k = 0; k < 4; ++k) {
      const v2f wg = sv[k] * (wslv[k] + walv[k] * (mc * ev[k]));  // B operand pair
      acc = __builtin_amdgcn_wmma_f32_16x16x4_f32(
          /*neg_a=*/false, A[k], /*neg_b=*/false, wg,
          /*c_mod=*/(short)0, acc, /*reuse_a=*/false, /*reuse_b=*/false);
    }

    // D lane layout: VGPR v -> out-channel (v + 8*half), pixel idx
    const size_t pbase = (size_t)(b * CN + 8 * half) * HWN + (size_t)(y * WW + x);
#pragma unroll
    for (int v = 0; v < 8; ++v) {
      const size_t off = pbase + (size_t)v * HWN;
      Qout[off] = unary[off] - acc[v];
    }
  }
}

// ---------------------------------------------------------------------------
extern "C" void kernel_launch(void* const* d_in, const int* in_sizes, int n_in,
                              void* d_out, int out_size, void* d_ws, size_t ws_size,
                              hipStream_t stream) {
  const float* unary    = (const float*)d_in[0];   // (4,16,64,1024)
  const float* xyz      = (const float*)d_in[1];   // (4,3,64,1024)
  const float* mask     = (const float*)d_in[2];   // (4,64,1024)
  const float* w_app    = (const float*)d_in[3];   // (1,16,1,1)
  const float* w_smooth = (const float*)d_in[4];   // (1,16,1,1)
  const float* Wc       = (const float*)d_in[5];   // (16,16)
  float* out   = (float*)d_out;
  float* kbeta = (float*)d_ws;                           // B*14*H*W floats
  float* wsQ   = kbeta + (size_t)BN * NTAP * HWN;        // B*C*H*W floats

  crf_beta_kernel<<<(BN * HWN) / 256, 256, 0, stream>>>(xyz, kbeta);

  dim3 grid(WW / TWID, HH / TH, BN);   // (16,16,4)
  // iter 1: unary -> out
  crf_iter_kernel<<<grid, 256, 0, stream>>>(unary, unary, mask, kbeta,
                                            w_app, w_smooth, Wc, out);
  // iter 2: out -> wsQ
  crf_iter_kernel<<<grid, 256, 0, stream>>>(out, unary, mask, kbeta,
                                            w_app, w_smooth, Wc, wsQ);
  // iter 3: wsQ -> out (final)
  crf_iter_kernel<<<grid, 256, 0, stream>>>(wsQ, unary, mask, kbeta,
                                            w_app, w_smooth, Wc, out);
}